// SetAbstraction_17617955848559
// MI455X (gfx1250) — compile-verified
//
#include <hip/hip_runtime.h>

typedef __attribute__((ext_vector_type(2))) float v2f;
typedef __attribute__((ext_vector_type(8))) float v8f;

#define Bn   4
#define Nn   8192
#define Dn   64
#define Sn   2048
#define Kn   32
#define R2c  0.04f      // RADIUS^2
#define GPB  4          // groups per block in MLP kernel
#define ST1  68         // LDS stride for h0 (67 ch padded to 68; conflict-free)
#define ST2  68         // LDS stride for a1 (64 ch padded)
#define ST3  132        // LDS stride for a2 (128 ch padded)

// f32 WMMA: D(16x16,f32) = A(16x4,f32) * B(4x16,f32) + C
// A frag per lane: lanes 0-15 hold A[M=lane][k0,k0+1], lanes 16-31 hold k0+2,k0+3
// B frag per lane: lanes 0-15 hold B[k0..k0+1][N=lane], lanes 16-31 hold rows k0+2,k0+3
__device__ __forceinline__ v8f wmma_f32_16x16x4(v2f a, v2f b, v8f c) {
  return __builtin_amdgcn_wmma_f32_16x16x4_f32(false, a, false, b, (short)0, c,
                                               false, false);
}

// ---------------------------------------------------------------------------
// Kernel 1: Farthest-point sampling. One block per batch, 1024 threads.
// Points + running min-distances live in registers (8 per thread).
// Writes new_xyz (= d_out[0 .. B*S*3)).
// ---------------------------------------------------------------------------
__global__ __launch_bounds__(1024) void fps_kernel(const float* __restrict__ xyz,
                                                   float* __restrict__ newxyz) {
  const int b = blockIdx.x;
  const int tid = threadIdx.x;
  const int lane = tid & 31;
  const int wid = tid >> 5;
  const float* xb = xyz + (size_t)b * (Nn * 3);

  float px[8], py[8], pz[8], pd[8];
#pragma unroll
  for (int j = 0; j < 8; ++j) {
    const int i = tid + (j << 10);
    px[j] = xb[i * 3 + 0];
    py[j] = xb[i * 3 + 1];
    pz[j] = xb[i * 3 + 2];
    pd[j] = 1e10f;
  }

  __shared__ float cpos[3];
  __shared__ float rv[32];
  __shared__ int   ri[32];
  __shared__ int   sFar;

  int far = 0;
  for (int s = 0; s < Sn; ++s) {
    // owner of point `far` publishes its coordinates
    if (tid == (far & 1023)) {
      const int j = far >> 10;
      cpos[0] = px[j]; cpos[1] = py[j]; cpos[2] = pz[j];
    }
    __syncthreads();
    const float cx = cpos[0], cy = cpos[1], cz = cpos[2];
    if (tid == 0) {
      float* o = newxyz + ((size_t)b * Sn + s) * 3;
      o[0] = cx; o[1] = cy; o[2] = cz;
    }
    // update distances, local argmax (strict > keeps lowest index per thread)
    float bv = -1.0f;
    int bi = 0;
#pragma unroll
    for (int j = 0; j < 8; ++j) {
      const float dx = px[j] - cx, dy = py[j] - cy, dz = pz[j] - cz;
      const float d = fminf(pd[j], dx * dx + dy * dy + dz * dz);
      pd[j] = d;
      const int i = tid + (j << 10);
      if (d > bv) { bv = d; bi = i; }
    }
    // wave32 argmax (tie -> lower index, matching jnp.argmax)
#pragma unroll
    for (int off = 16; off > 0; off >>= 1) {
      const float ov = __shfl_down(bv, off, 32);
      const int   oi = __shfl_down(bi, off, 32);
      if (ov > bv || (ov == bv && oi < bi)) { bv = ov; bi = oi; }
    }
    if (lane == 0) { rv[wid] = bv; ri[wid] = bi; }
    __syncthreads();
    if (wid == 0) {
      float v = rv[lane];
      int ii = ri[lane];
#pragma unroll
      for (int off = 16; off > 0; off >>= 1) {
        const float ov = __shfl_down(v, off, 32);
        const int   oi = __shfl_down(ii, off, 32);
        if (ov > v || (ov == v && oi < ii)) { v = ov; ii = oi; }
      }
      if (lane == 0) sFar = ii;
    }
    __syncthreads();
    far = sFar;
  }
}

// ---------------------------------------------------------------------------
// Kernel 2: ball query. One wave per centroid; first K in-order hits within
// radius via ballot/rank compaction; pad tail with first hit.
// ---------------------------------------------------------------------------
__global__ __launch_bounds__(256) void ball_query_kernel(
    const float* __restrict__ xyz, const float* __restrict__ newxyz,
    int* __restrict__ grp) {
  const int lane = threadIdx.x & 31;
  const int wid = threadIdx.x >> 5;
  const int gidx = blockIdx.x * 8 + wid;      // 0 .. B*S-1
  const int b = gidx >> 11;                   // S = 2048
  const float* xb = xyz + (size_t)b * (Nn * 3);
  const float cx = newxyz[gidx * 3 + 0];
  const float cy = newxyz[gidx * 3 + 1];
  const float cz = newxyz[gidx * 3 + 2];
  int* g = grp + (size_t)gidx * Kn;

  int cnt = 0;       // wave-uniform
  int first = -1;    // wave-uniform
  for (int n0 = 0; n0 < Nn && cnt < Kn; n0 += 32) {
    const int n = n0 + lane;
    const float dx = xb[n * 3 + 0] - cx;
    const float dy = xb[n * 3 + 1] - cy;
    const float dz = xb[n * 3 + 2] - cz;
    const bool in = (dx * dx + dy * dy + dz * dz) <= R2c;
    const unsigned mask = (unsigned)__ballot(in);
    if (in) {
      const int rank = __popc(mask & ((1u << lane) - 1u));
      const int pos = cnt + rank;
      if (pos < Kn) g[pos] = n;
    }
    if (first < 0 && mask) first = n0 + __ffs(mask) - 1;
    cnt += __popc(mask);
  }
  if (first < 0) first = 0;
  if (cnt > Kn) cnt = Kn;
  for (int j = cnt + lane; j < Kn; j += 32) g[j] = first;
}

// ---------------------------------------------------------------------------
// Kernel 3: gather + 3-layer MLP (f32 WMMA 16x16x4) + max over K.
// One block = GPB groups = GPB*32 points, 128 threads = 4 waves.
// Each wave owns (group, 16-output-column) strips; inside a strip it K-loops
// WMMAs for both 16-point M-tiles of the group, reusing the B (weight) frags.
// ---------------------------------------------------------------------------
template <int CIN, int KP, int SIN, int COUT, int SOUT, bool GUARD>
__device__ __forceinline__ void mlp_layer(const float* __restrict__ W,
                                          const float* __restrict__ bias,
                                          const float* __restrict__ aIn,
                                          float* __restrict__ aOut,
                                          int lane, int wid) {
  const int NT = COUT >> 4;
  const int l16 = lane & 15;
  const bool hi = lane >= 16;
  const int koff = hi ? 2 : 0;
  for (int t = wid; t < GPB * NT; t += 4) {
    const int g = t & (GPB - 1);
    const int nb = (t / GPB) << 4;
    const int wrow = nb + l16;
    const float* Wr = W + (size_t)wrow * CIN;
    const float* A0 = aIn + (g * 32 + l16) * SIN;
    const float* A1 = A0 + 16 * SIN;
    v8f acc0 = {}, acc1 = {};
    for (int cb = 0; cb < KP; cb += 4) {
      const int c0 = cb + koff;
      v2f bf, af0, af1;
      if (GUARD) {
        bf.x = (c0 < CIN) ? Wr[c0] : 0.0f;
        bf.y = (c0 + 1 < CIN) ? Wr[c0 + 1] : 0.0f;
      } else {
        bf.x = Wr[c0];
        bf.y = Wr[c0 + 1];
      }
      af0.x = A0[c0]; af0.y = A0[c0 + 1];
      af1.x = A1[c0]; af1.y = A1[c0 + 1];
      acc0 = wmma_f32_16x16x4(af0, bf, acc0);
      acc1 = wmma_f32_16x16x4(af1, bf, acc1);
    }
    const float bv = bias[wrow];
    const int roff = hi ? 8 : 0;  // C VGPR i -> rows M=i (lanes<16), M=i+8
    float* O0 = aOut + (g * 32 + roff) * SOUT + wrow - nb + nb;  // = +nb+l16
#pragma unroll
    for (int i = 0; i < 8; ++i) O0[i * SOUT] = fmaxf(acc0[i] + bv, 0.0f);
    float* O1 = O0 + 16 * SOUT;
#pragma unroll
    for (int i = 0; i < 8; ++i) O1[i * SOUT] = fmaxf(acc1[i] + bv, 0.0f);
  }
}

__device__ __forceinline__ void mlp_layer3_max(const float* __restrict__ W,
                                               const float* __restrict__ bias,
                                               const float* __restrict__ aIn,
                                               float* __restrict__ out,
                                               int gbase, int lane, int wid) {
  const int l16 = lane & 15;
  const bool hi = lane >= 16;
  const int koff = hi ? 2 : 0;
  for (int t = wid; t < GPB * 16; t += 4) {
    const int g = t & (GPB - 1);
    const int nb = (t / GPB) << 4;
    const int wrow = nb + l16;
    const float* Wr = W + (size_t)wrow * 128;
    const float* A0 = aIn + (g * 32 + l16) * ST3;
    const float* A1 = A0 + 16 * ST3;
    v8f acc0 = {}, acc1 = {};
    for (int cb = 0; cb < 128; cb += 4) {
      const int c0 = cb + koff;
      v2f bf, af0, af1;
      bf.x = Wr[c0];  bf.y = Wr[c0 + 1];
      af0.x = A0[c0]; af0.y = A0[c0 + 1];
      af1.x = A1[c0]; af1.y = A1[c0 + 1];
      acc0 = wmma_f32_16x16x4(af0, bf, acc0);
      acc1 = wmma_f32_16x16x4(af1, bf, acc1);
    }
    // max over the group's 32 points: 16 register values + one shfl_xor(16).
    // relu & bias commute with max (relu monotone, bias uniform over points).
    float pm = fmaxf(acc0[0], acc1[0]);
#pragma unroll
    for (int i = 1; i < 8; ++i) pm = fmaxf(pm, fmaxf(acc0[i], acc1[i]));
    pm = fmaxf(pm, __shfl_xor(pm, 16, 32));
    const float v = fmaxf(pm + bias[wrow], 0.0f);
    if (!hi) out[(size_t)(gbase + g) * 256 + wrow] = v;
  }
}

__global__ __launch_bounds__(128) void group_mlp_kernel(
    const float* __restrict__ xyz, const float* __restrict__ features,
    const float* __restrict__ newxyz, const int* __restrict__ grp,
    const float* __restrict__ W1, const float* __restrict__ b1,
    const float* __restrict__ W2, const float* __restrict__ b2,
    const float* __restrict__ W3, const float* __restrict__ b3,
    float* __restrict__ feats_out) {
  __shared__ float h0[GPB * 32 * ST1];   // [128][68]  rel-xyz(3)+feat(64)+pad
  __shared__ float a1[GPB * 32 * ST2];   // [128][68]
  __shared__ float a2[GPB * 32 * ST3];   // [128][132]
  const int tid = threadIdx.x;
  const int lane = tid & 31;
  const int wid = tid >> 5;
  const int gbase = blockIdx.x * GPB;

  // ---- gather: one thread per point ----
  {
    const int g = tid >> 5, k = tid & 31;
    const int gidx = gbase + g;
    const int b = gidx >> 11;  // S = 2048
    const int nidx = grp[(size_t)gidx * Kn + k];
    const float* pb = xyz + ((size_t)b * Nn + nidx) * 3;
    const float* cb = newxyz + (size_t)gidx * 3;
    float* hrow = h0 + tid * ST1;
    hrow[0] = pb[0] - cb[0];
    hrow[1] = pb[1] - cb[1];
    hrow[2] = pb[2] - cb[2];
    const float* fb = features + ((size_t)b * Nn + nidx) * Dn;
#pragma unroll 8
    for (int c = 0; c < Dn; ++c) hrow[3 + c] = fb[c];
    hrow[67] = 0.0f;
  }
  __syncthreads();

  mlp_layer<67, 68, ST1, 64, ST2, true>(W1, b1, h0, a1, lane, wid);
  __syncthreads();
  mlp_layer<64, 64, ST2, 128, ST3, false>(W2, b2, a1, a2, lane, wid);
  __syncthreads();
  mlp_layer3_max(W3, b3, a2, feats_out, gbase, lane, wid);
}

// ---------------------------------------------------------------------------
// Launcher. d_in order: xyz, features, W1, b1, W2, b2, W3, b3.
// d_out: new_xyz (B*S*3 floats) then feats (B*S*256 floats).
// d_ws: grp_idx (B*S*K int32 = 1 MiB).
// ---------------------------------------------------------------------------
extern "C" void kernel_launch(void* const* d_in, const int* in_sizes, int n_in,
                              void* d_out, int out_size, void* d_ws,
                              size_t ws_size, hipStream_t stream) {
  (void)in_sizes; (void)n_in; (void)out_size; (void)ws_size;
  const float* xyz      = (const float*)d_in[0];
  const float* features = (const float*)d_in[1];
  const float* W1 = (const float*)d_in[2];
  const float* b1 = (const float*)d_in[3];
  const float* W2 = (const float*)d_in[4];
  const float* b2 = (const float*)d_in[5];
  const float* W3 = (const float*)d_in[6];
  const float* b3 = (const float*)d_in[7];

  float* out = (float*)d_out;
  float* newxyz = out;                              // B*S*3
  float* feats  = out + (size_t)Bn * Sn * 3;        // B*S*256
  int* grp = (int*)d_ws;                            // B*S*K ints

  fps_kernel<<<Bn, 1024, 0, stream>>>(xyz, newxyz);
  ball_query_kernel<<<(Bn * Sn) / 8, 256, 0, stream>>>(xyz, newxyz, grp);
  group_mlp_kernel<<<(Bn * Sn) / GPB, 128, 0, stream>>>(
      xyz, features, newxyz, grp, W1, b1, W2, b2, W3, b3, feats);
}